// QGN_30219389895238
// MI455X (gfx1250) — compile-verified
//
#include <hip/hip_runtime.h>

// Quantum Rot-gate chain, elementwise over B = 4,194,304 rows of 12 floats.
// Memory-bound: ~218 MB traffic -> ~9.4 us floor at 23.3 TB/s HBM.
// Hardware v_sin_f32/v_cos_f32 (via __sincosf) keep compute under that floor.

#define QGN_BLOCK 256
#define QGN_PF_ROWS 2048   // prefetch distance in rows (~96 KB ahead, lives in 192MB L2)

__device__ __forceinline__ void rot_gate(float& s0r, float& s0i,
                                         float& s1r, float& s1i,
                                         float phi, float theta, float omega) {
  // Rot(phi,theta,omega) = RZ(omega) RY(theta) RZ(phi)
  // m00 = e^{-i(phi+omega)/2} c   m01 = -e^{ i(phi-omega)/2} s
  // m10 = e^{-i(phi-omega)/2} s   m11 =  e^{ i(phi+omega)/2} c
  float sh, ch;
  __sincosf(0.5f * theta, &sh, &ch);          // c = cos(theta/2), s = sin(theta/2)
  const float a = 0.5f * (phi + omega);
  const float b = 0.5f * (phi - omega);
  float sa, ca, sb, cb;
  __sincosf(a, &sa, &ca);
  __sincosf(b, &sb, &cb);

  const float m00r =  ca * ch, m00i = -sa * ch;   // e^{-ia} * c
  const float m01r = -cb * sh, m01i = -sb * sh;   // -e^{ib} * s
  const float m10r =  cb * sh, m10i = -sb * sh;   // conj(e^{ib}) * s
  const float m11r =  ca * ch, m11i =  sa * ch;   // e^{ia} * c

  const float n0r = m00r * s0r - m00i * s0i + m01r * s1r - m01i * s1i;
  const float n0i = m00r * s0i + m00i * s0r + m01r * s1i + m01i * s1r;
  const float n1r = m10r * s0r - m10i * s0i + m11r * s1r - m11i * s1i;
  const float n1i = m10r * s0i + m10i * s0r + m11r * s1i + m11i * s1r;
  s0r = n0r; s0i = n0i; s1r = n1r; s1i = n1i;
}

__global__ __launch_bounds__(QGN_BLOCK) void qgn_rot_chain_kernel(
    const float* __restrict__ x,   // [n, 12]
    const float* __restrict__ w,   // [24], uniform -> scalar loads
    float* __restrict__ out,       // [n]
    int n) {
  const int i = (int)(blockIdx.x * QGN_BLOCK + threadIdx.x);
  if (i >= n) return;

  // CDNA5 global_prefetch_b8: stream a future row's cacheline toward L2.
  // Clamped in-bounds so the access is defined either way.
  {
    const int pf = (i + QGN_PF_ROWS < n) ? (i + QGN_PF_ROWS) : i;
    __builtin_prefetch(x + (size_t)pf * 12, 0, 0);
  }

  // Row i = 12 contiguous floats = 3 x b128 loads. A wave's three b128 loads
  // jointly cover a contiguous 1536 B span -> full cacheline utilization.
  const float4* __restrict__ xv = reinterpret_cast<const float4*>(x) + (size_t)i * 3;
  const float4 x0 = xv[0];   // x[0..3]
  const float4 x1 = xv[1];   // x[4..7]
  const float4 x2 = xv[2];   // x[8..11]

  // Initial state |0>: compiler constant-folds gate 0 against (1,0),(0,0).
  float s0r = 1.0f, s0i = 0.0f, s1r = 0.0f, s1i = 0.0f;

  rot_gate(s0r, s0i, s1r, s1i,
           fmaf(x0.x, w[1],  w[0]),
           fmaf(x0.y, w[3],  w[2]),
           fmaf(x0.z, w[5],  w[4]));
  rot_gate(s0r, s0i, s1r, s1i,
           fmaf(x0.w, w[7],  w[6]),
           fmaf(x1.x, w[9],  w[8]),
           fmaf(x1.y, w[11], w[10]));
  rot_gate(s0r, s0i, s1r, s1i,
           fmaf(x1.z, w[13], w[12]),
           fmaf(x1.w, w[15], w[14]),
           fmaf(x2.x, w[17], w[16]));
  rot_gate(s0r, s0i, s1r, s1i,
           fmaf(x2.y, w[19], w[18]),
           fmaf(x2.z, w[21], w[20]),
           fmaf(x2.w, w[23], w[22]));

  out[i] = fmaf(s0r, s0r, s0i * s0i);   // |<0|psi>|^2
}

extern "C" void kernel_launch(void* const* d_in, const int* in_sizes, int n_in,
                              void* d_out, int out_size, void* d_ws, size_t ws_size,
                              hipStream_t stream) {
  (void)n_in; (void)out_size; (void)d_ws; (void)ws_size;
  const float* x = (const float*)d_in[0];   // [B,12] fp32
  const float* w = (const float*)d_in[1];   // [24]   fp32
  float* out = (float*)d_out;               // [B]    fp32

  const int n = in_sizes[0] / 12;           // B
  const int grid = (n + QGN_BLOCK - 1) / QGN_BLOCK;
  qgn_rot_chain_kernel<<<grid, QGN_BLOCK, 0, stream>>>(x, w, out, n);
}